// LapLoss_72456098283940
// MI455X (gfx1250) — compile-verified
//
#include <hip/hip_runtime.h>

// ---------------------------------------------------------------------------
// Laplacian pyramid L1 loss, fused for MI455X (gfx1250, wave32).
// Bandwidth-bound (~0.4 GB total traffic -> ~18us at 23.3 TB/s), so the
// plan is LDS-tiled f32 compute; the CDNA5 matrix pipe is used for the
// wave-level reduction (V_WMMA_F32_16X16X4_F32 with an all-ones B matrix,
// which makes the result independent of the B VGPR layout).
// ---------------------------------------------------------------------------

#define BB   16
#define CCH  3
#define NIMG (BB * CCH)

typedef float v2f __attribute__((ext_vector_type(2)));
typedef float v8f __attribute__((ext_vector_type(8)));

__device__ __forceinline__ int reflect(int i, int n) {
    // jnp.pad 'reflect' (no edge repeat); valid for |overhang| <= n-2
    if (i < 0)  i = -i;
    if (i >= n) i = 2 * n - 2 - i;
    return i;
}

// Wave32 sum of one float per lane using the matrix pipe.
// Lane L contributes A[M=L&15][K=2*(L>>4)]; B = ones(4x16) so
// D[m][n] = sum_k A[m][k] regardless of B's lane layout. Lane L then holds
// D rows {0..7}+8*(L>>4) at column L&15; summing its 8 values gives a
// half-sum, and xor-16 shuffle combines the two halves.
__device__ __forceinline__ float wave_reduce_wmma(float v) {
    v2f a;  a.x = v;    a.y = 0.0f;
    v2f b;  b.x = 1.0f; b.y = 1.0f;
    v8f c = {};
    v8f d = __builtin_amdgcn_wmma_f32_16x16x4_f32(
        /*neg_a=*/false, a, /*neg_b=*/false, b,
        /*c_mod=*/(short)0, c, /*reuse_a=*/false, /*reuse_b=*/false);
    float hs = d[0] + d[1] + d[2] + d[3] + d[4] + d[5] + d[6] + d[7];
    return hs + __shfl_xor(hs, 16, 32);
}

__global__ void zero_out_kernel(float* out) {
    if (threadIdx.x == 0) out[0] = 0.0f;
}

// ---------------------------------------------------------------------------
// kdown: gauss5x5(reflect-pad-2) then stride-2 downsample, for both tensors.
// One 16x16 output tile per block; 35x35 input tile staged in LDS.
// ---------------------------------------------------------------------------
__global__ __launch_bounds__(256) void kdown(
    const float* __restrict__ curIn, const float* __restrict__ curTg,
    const float* __restrict__ gk,   // (3,1,5,5)
    float* __restrict__ dIn, float* __restrict__ dTg,
    int H, int W)
{
    const int Hd = H >> 1, Wd = W >> 1;
    const int img = blockIdx.z;
    const int c   = img % CCH;
    const int i0  = blockIdx.y * 16;
    const int j0  = blockIdx.x * 16;
    const int tid = threadIdx.y * 16 + threadIdx.x;

    __shared__ float sIn[35][36];
    __shared__ float sTg[35][36];
    __shared__ float sK[25];

    if (tid < 25) sK[tid] = gk[c * 25 + tid];

    const float* baseIn = curIn + (size_t)img * H * W;
    const float* baseTg = curTg + (size_t)img * H * W;
    const int r0 = 2 * i0 - 2, q0 = 2 * j0 - 2;
    for (int idx = tid; idx < 35 * 35; idx += 256) {
        int rr = idx / 35, cc = idx - rr * 35;
        int gr = reflect(r0 + rr, H);
        int gc = reflect(q0 + cc, W);
        size_t off = (size_t)gr * W + gc;
        sIn[rr][cc] = baseIn[off];
        sTg[rr][cc] = baseTg[off];
    }
    __syncthreads();

    const int i = i0 + threadIdx.y, j = j0 + threadIdx.x;
    if (i < Hd && j < Wd) {
        float aI = 0.0f, aT = 0.0f;
        const int ly = 2 * threadIdx.y, lx = 2 * threadIdx.x;
#pragma unroll
        for (int a = 0; a < 5; ++a)
#pragma unroll
            for (int b = 0; b < 5; ++b) {
                float w = sK[a * 5 + b];
                aI += w * sIn[ly + a][lx + b];
                aT += w * sTg[ly + a][lx + b];
            }
        size_t o = (size_t)img * Hd * Wd + (size_t)i * Wd + j;
        dIn[o] = aI;
        dTg[o] = aT;
    }
}

// ---------------------------------------------------------------------------
// kloss: build diff = cur - upsample(down) tile (18x18, both tensors) in LDS,
// then |sobelx|+|sobely| on the pad-2 grid (H+2 x W+2), reduce
// |ml_in - ml_tg| * invCnt into the global loss scalar.
// ---------------------------------------------------------------------------
__global__ __launch_bounds__(256) void kloss(
    const float* __restrict__ curIn, const float* __restrict__ curTg,
    const float* __restrict__ dIn,  const float* __restrict__ dTg,
    const float* __restrict__ gk,
    float* __restrict__ out,
    int H, int W, float invCnt)
{
    const int HP = H + 2, WP = W + 2;
    const int Hd = H >> 1, Wd = W >> 1;
    const int img = blockIdx.z;
    const int c   = img % CCH;
    const int y0  = blockIdx.y * 16;
    const int x0  = blockIdx.x * 16;
    const int tid = threadIdx.y * 16 + threadIdx.x;

    __shared__ float sdI[18][19];
    __shared__ float sdT[18][19];
    __shared__ float sUK[25];   // 4 * gauss kernel (upsample weights)
    __shared__ float red[8];

    if (tid < 25) sUK[tid] = 4.0f * gk[c * 25 + tid];
    __syncthreads();

    const float* baseI = curIn + (size_t)img * H * W;
    const float* baseT = curTg + (size_t)img * H * W;
    const float* bdI   = dIn  + (size_t)img * Hd * Wd;
    const float* bdT   = dTg  + (size_t)img * Hd * Wd;

    // diffpad[p][q] = diff[reflect(p-2,H)][reflect(q-2,W)]; local s,t cover
    // p = y0+s, q = x0+t for s,t in [0,18).
    for (int idx = tid; idx < 18 * 18; idx += 256) {
        int s = idx / 18, t = idx - s * 18;
        int u = reflect(y0 - 2 + s, H);
        int v = reflect(x0 - 2 + t, W);

        // up[u][v]: zero-interleaved down, reflect-pad-2 on the H-grid,
        // conv with 4*gk. Nonzero taps where (u+dy) and (v+dx) are even;
        // reflect on an even-size grid preserves parity.
        float upI = 0.0f, upT = 0.0f;
#pragma unroll
        for (int dy = -2; dy <= 2; ++dy) {
            int m = u + dy;
            if ((m & 1) == 0) {
                int md = reflect(m, H) >> 1;
#pragma unroll
                for (int dx = -2; dx <= 2; ++dx) {
                    int n = v + dx;
                    if ((n & 1) == 0) {
                        int nd = reflect(n, W) >> 1;
                        float w = sUK[(dy + 2) * 5 + (dx + 2)];
                        size_t o = (size_t)md * Wd + nd;
                        upI += w * bdI[o];
                        upT += w * bdT[o];
                    }
                }
            }
        }
        size_t o = (size_t)u * W + v;
        sdI[s][t] = baseI[o] - upI;
        sdT[s][t] = baseT[o] - upT;
    }
    __syncthreads();

    float contrib = 0.0f;
    const int y = y0 + threadIdx.y, x = x0 + threadIdx.x;
    if (y < HP && x < WP) {
        const int s = threadIdx.y, t = threadIdx.x;
        // GX = [[2,0,-2],[4,0,-4],[2,0,-2]], GY = [[2,4,2],[0,0,0],[-2,-4,-2]]
        float gxI = 2.f*(sdI[s  ][t] - sdI[s  ][t+2])
                  + 4.f*(sdI[s+1][t] - sdI[s+1][t+2])
                  + 2.f*(sdI[s+2][t] - sdI[s+2][t+2]);
        float gyI = 2.f*(sdI[s][t  ] - sdI[s+2][t  ])
                  + 4.f*(sdI[s][t+1] - sdI[s+2][t+1])
                  + 2.f*(sdI[s][t+2] - sdI[s+2][t+2]);
        float gxT = 2.f*(sdT[s  ][t] - sdT[s  ][t+2])
                  + 4.f*(sdT[s+1][t] - sdT[s+1][t+2])
                  + 2.f*(sdT[s+2][t] - sdT[s+2][t+2]);
        float gyT = 2.f*(sdT[s][t  ] - sdT[s+2][t  ])
                  + 4.f*(sdT[s][t+1] - sdT[s+2][t+1])
                  + 2.f*(sdT[s][t+2] - sdT[s+2][t+2]);
        float mlI = fabsf(gxI) + fabsf(gyI);
        float mlT = fabsf(gxT) + fabsf(gyT);
        contrib = fabsf(mlI - mlT) * invCnt;
    }

    // Reduction: matrix-pipe wave reduce (EXEC is all-ones here: OOB threads
    // contribute 0 by value, not by masking), then LDS across 8 waves.
    float wsum = wave_reduce_wmma(contrib);
    const int lane = tid & 31, wid = tid >> 5;
    if (lane == 0) red[wid] = wsum;
    __syncthreads();
    if (tid == 0) {
        float tot = 0.0f;
#pragma unroll
        for (int k = 0; k < 8; ++k) tot += red[k];
        atomicAdd(out, tot);
    }
}

// ---------------------------------------------------------------------------
// Host-side orchestration (graph-capture safe: stream-only, deterministic).
// ---------------------------------------------------------------------------
extern "C" void kernel_launch(void* const* d_in, const int* in_sizes, int n_in,
                              void* d_out, int out_size, void* d_ws, size_t ws_size,
                              hipStream_t stream)
{
    (void)in_sizes; (void)n_in; (void)out_size; (void)ws_size;

    const float* inp = (const float*)d_in[0];
    const float* tgt = (const float*)d_in[1];
    const float* gk  = (const float*)d_in[2];
    float* out = (float*)d_out;
    float* ws  = (float*)d_ws;

    // Workspace: down pyramids for both tensors (~33 MB total).
    const size_t N1 = (size_t)NIMG * 256 * 256;
    const size_t N2 = (size_t)NIMG * 128 * 128;
    const size_t N3 = (size_t)NIMG * 64 * 64;
    float* dIn1 = ws;          float* dTg1 = dIn1 + N1;
    float* dIn2 = dTg1 + N1;   float* dTg2 = dIn2 + N2;
    float* dIn3 = dTg2 + N2;   float* dTg3 = dIn3 + N3;

    zero_out_kernel<<<1, 64, 0, stream>>>(out);

    const dim3 blk(16, 16);

    // Level 0: H=512
    kdown<<<dim3(16, 16, NIMG), blk, 0, stream>>>(inp, tgt, gk, dIn1, dTg1, 512, 512);
    kloss<<<dim3(33, 33, NIMG), blk, 0, stream>>>(inp, tgt, dIn1, dTg1, gk, out,
                                                  512, 512, 1.0f / ((float)NIMG * 514.0f * 514.0f));
    // Level 1: H=256, current = down level-0
    kdown<<<dim3(8, 8, NIMG), blk, 0, stream>>>(dIn1, dTg1, gk, dIn2, dTg2, 256, 256);
    kloss<<<dim3(17, 17, NIMG), blk, 0, stream>>>(dIn1, dTg1, dIn2, dTg2, gk, out,
                                                  256, 256, 1.0f / ((float)NIMG * 258.0f * 258.0f));
    // Level 2: H=128, current = down level-1
    kdown<<<dim3(4, 4, NIMG), blk, 0, stream>>>(dIn2, dTg2, gk, dIn3, dTg3, 128, 128);
    kloss<<<dim3(9, 9, NIMG), blk, 0, stream>>>(dIn2, dTg2, dIn3, dTg3, gk, out,
                                                128, 128, 1.0f / ((float)NIMG * 130.0f * 130.0f));
}